// Quantizer_65824668779018
// MI455X (gfx1250) — compile-verified
//
#include <hip/hip_runtime.h>
#include <math.h>

typedef float v2f __attribute__((ext_vector_type(2)));
typedef float v4f __attribute__((ext_vector_type(4)));
typedef float v8f __attribute__((ext_vector_type(8)));

#define N_ROWS        65536
#define E_DIM         1024
#define K_STAGES      4
#define D_DIM         5
#define RSTRIDE       1028          // padded LDS row stride (floats): bank-conflict-free patterns
#define WAVES_PER_WG  2
#define ROWS_PER_WAVE 16
#define N_TILES       (N_ROWS / ROWS_PER_WAVE)            // 4096 waves total
#define LDS_BYTES     (WAVES_PER_WG * ROWS_PER_WAVE * RSTRIDE * 4)

#define Q_ELEMS    ((size_t)N_ROWS * E_DIM)
#define CODE_ELEMS ((size_t)N_ROWS * K_STAGES * D_DIM)

__global__ __launch_bounds__(WAVES_PER_WG * 32, 1)
void rvq_fused(const float* __restrict__ h_in,
               const float* __restrict__ Wp,   // (K,D,E)
               const float* __restrict__ bp,   // (K,D)
               const float* __restrict__ Wi,   // (K,E,D)
               const float* __restrict__ bi,   // (K,E)
               const float* __restrict__ temp, // (1,)
               const float* __restrict__ u,    // (K,N,D,C)
               const float* __restrict__ p,    // (N,)
               float* __restrict__ out_q,
               float* __restrict__ out_code,
               float* __restrict__ ws)
{
  extern __shared__ __align__(16) float lds[];
  const int tid  = threadIdx.x;
  const int lane = tid & 31;
  const int wave = tid >> 5;
  const int half = lane >> 4;         // 0: lanes 0-15, 1: lanes 16-31
  const int row  = lane & 15;
  const int tile = blockIdx.x * WAVES_PER_WG + wave;
  const int row0 = tile * ROWS_PER_WAVE;
  const int koff = half * 2;

  float* rbuf = lds + wave * (ROWS_PER_WAVE * RSTRIDE);

  // ---- constants derived exactly like numpy float64 path, then cast to f32 ----
  const double hl0d = 7.0 * (1.0 - 1e-3) * 0.5;   // dims 0..2 (levels 8)
  const double hl3d = 5.0 * (1.0 - 1e-3) * 0.5;   // dim 3 (level 6)
  const double hl4d = 4.0 * (1.0 - 1e-3) * 0.5;   // dim 4 (level 5)
  const float HL0 = (float)hl0d, HL3 = (float)hl3d, HL4 = (float)hl4d;
  const float SH0 = (float)tan(0.5 / hl0d);
  const float SH3 = (float)tan(0.5 / hl3d);
  const float SH4 = 0.0f;

  const float tv  = temp[0];
  const float te  = fmaxf(tv * tv, 1e-8f);
  const float am1 = 1.0f / te - 1.0f;             // alpha - 1

  // ---- selected stage for the loss mixture (i/4 < p <= (i+1)/4) ----
  const float pv = p[row0 + row];
  int isel = -1;
#pragma unroll
  for (int i = 0; i < K_STAGES; ++i) {
    if ((i * 0.25f < pv) && (pv <= (i + 1) * 0.25f)) isel = i;
  }

  // ---- stream h_in tile into LDS residual buffer ----
  for (int r = 0; r < ROWS_PER_WAVE; ++r) {
    const float* src = h_in + (size_t)(row0 + r) * E_DIM;
    float* dst = rbuf + r * RSTRIDE;
#pragma unroll
    for (int c = 0; c < 8; ++c) {
      const int e = c * 128 + lane * 4;
      *(v4f*)(dst + e) = *(const v4f*)(src + e);
    }
  }

  float rowssq = 0.0f;  // lane n (<16) ends up holding ||r_snapshot(row n)||^2

  auto snap = [&](int s) {
    for (int r = 0; r < ROWS_PER_WAVE; ++r) {
      const int rs = __shfl(isel, r, 32);
      if (rs != s) continue;
      const float* src = rbuf + r * RSTRIDE;
      float a = 0.0f;
#pragma unroll
      for (int c = 0; c < 8; ++c) {
        v4f v = *(const v4f*)(src + c * 128 + lane * 4);
        a = fmaf(v[0], v[0], a); a = fmaf(v[1], v[1], a);
        a = fmaf(v[2], v[2], a); a = fmaf(v[3], v[3], a);
      }
      a += __shfl_xor(a, 1, 32);  a += __shfl_xor(a, 2, 32);
      a += __shfl_xor(a, 4, 32);  a += __shfl_xor(a, 8, 32);
      a += __shfl_xor(a, 16, 32);
      if (lane == r) rowssq = a;
    }
  };

  snap(-1);  // rows with p <= 0: loss uses ||h_in||

  const int dd = (row < D_DIM) ? row : 0;  // pad A-matrix rows 5..15 harmlessly

#pragma unroll 1
  for (int i = 0; i < K_STAGES; ++i) {
    const float* Wpi = Wp + (size_t)i * D_DIM * E_DIM;
    const float* bpi = bp + i * D_DIM;
    const float* Wii = Wi + (size_t)i * E_DIM * D_DIM;
    const float* bii = bi + (size_t)i * E_DIM;

    // ---------- projection: h^T(16x16) = Wp_chunk(16x4) x r^T_chunk(4x16), K-loop over E ----------
    v8f a0 = {0,0,0,0,0,0,0,0}, a1 = a0, a2 = a0, a3 = a0;
    const float* wrow = Wpi + (size_t)dd * E_DIM + koff;  // A: lane = d-row, K pair (koff,koff+1)
    const float* rrow = rbuf + row * RSTRIDE + koff;      // B: lane = batch-row, K pair
    for (int k = 0; k < 64; ++k) {
      const int e0 = k * 16;
      v2f A, B;
      A = *(const v2f*)(wrow + e0);
      B = *(const v2f*)(rrow + e0);
      a0 = __builtin_amdgcn_wmma_f32_16x16x4_f32(false, A, false, B, (short)0, a0, false, false);
      A = *(const v2f*)(wrow + e0 + 4);
      B = *(const v2f*)(rrow + e0 + 4);
      a1 = __builtin_amdgcn_wmma_f32_16x16x4_f32(false, A, false, B, (short)0, a1, false, false);
      A = *(const v2f*)(wrow + e0 + 8);
      B = *(const v2f*)(rrow + e0 + 8);
      a2 = __builtin_amdgcn_wmma_f32_16x16x4_f32(false, A, false, B, (short)0, a2, false, false);
      A = *(const v2f*)(wrow + e0 + 12);
      B = *(const v2f*)(rrow + e0 + 12);
      a3 = __builtin_amdgcn_wmma_f32_16x16x4_f32(false, A, false, B, (short)0, a3, false, false);
    }
    v8f hsum = (a0 + a1) + (a2 + a3);  // lane n(<16): hsum[d] = h[d][row n] (pre-bias)

    const float bp0 = bpi[0], bp1 = bpi[1], bp2 = bpi[2], bp3 = bpi[3], bp4 = bpi[4];
    // ship h[3],h[4] to the upper half-wave
    const float h3v = __shfl_xor(hsum[3], 16, 32) + bp3;  // valid in lanes 16-31
    const float h4v = __shfl_xor(hsum[4], 16, 32) + bp4;

    // ---------- FSQ + gumbel-argmax quantize: low half handles d=0,1,2; high half d=3,4 ----------
    const float* ub = u + ((size_t)i * N_ROWS + (size_t)(row0 + row)) * (D_DIM * 8);
    float zA = 0.0f, zB = 0.0f, zC = 0.0f;
#pragma unroll
    for (int s = 0; s < 3; ++s) {
      const int d = half ? (3 + s) : s;
      const bool active = (d < D_DIM);
      const float hq  = half ? (s == 0 ? h3v : h4v)
                             : hsum[s] + (s == 0 ? bp0 : (s == 1 ? bp1 : bp2));
      const float cHL = half ? (s == 0 ? HL3 : HL4) : HL0;
      const float cSH = half ? (s == 0 ? SH3 : SH4) : SH0;
      const float cOF = (half && s >= 1) ? 0.0f : 0.5f;
      const int   lo  = half ? (s == 0 ? -3 : -2) : -4;
      const int   hi  = half ? 2 : 3;
      const float cHW = half ? (s == 0 ? 3.0f : 2.0f) : 4.0f;

      float zq = 0.0f;
      if (active) {
        const float z = tanhf(hq + cSH) * cHL - cOF;
        const v4f u0 = *(const v4f*)(ub + d * 8);
        const v4f u1 = *(const v4f*)(ub + d * 8 + 4);
        float best = -3.0e38f, bstep = 0.0f;
#pragma unroll
        for (int cc = 0; cc < 8; ++cc) {
          const int st = cc - 4;
          const float uv = (cc < 4) ? u0[cc] : u1[cc - 4];
          const float g  = -logf(-logf(uv + 1e-20f) + 1e-20f);
          const float df = z - (float)st;
          const float d2 = df * df;
          const float d2m = am1 * d2 + d2;               // temperature trick (forward value)
          const float sc = -(d2m / te) + g;
          const bool take = (st >= lo) && (st <= hi) && (sc > best);  // first-max wins
          best  = take ? sc : best;
          bstep = take ? (float)st : bstep;
        }
        zq = bstep / cHW;
        out_code[(size_t)(row0 + row) * (K_STAGES * D_DIM) + i * D_DIM + d] = zq;
      }
      if (s == 0) zA = zq; else if (s == 1) zB = zq; else zC = zq;
    }

    // ---------- build B = z^T (4x16) operands; bias folded in at K=5 ----------
    const float t2 = __shfl_xor(zC, 16, 32);  // high lanes get low's z2
    const float t4 = __shfl_xor(zB, 16, 32);  // low lanes get high's z4
    v2f B0, B1;
    B0[0] = half ? t2 : zA;      // K0: z0   | K2: z2
    B0[1] = half ? zA : zB;      // K1: z1   | K3: z3
    B1[0] = half ? 0.0f : t4;    // K4: z4   | K6: 0
    B1[1] = half ? 0.0f : 1.0f;  // K5: bias | K7: 0

    // ---------- inverse GEMM + residual update: q^T(16e x 16row) chunks over E ----------
    for (int c = 0; c < 64; ++c) {
      const int e = c * 16 + row;                 // A row (M) = e-index
      const float* wie = Wii + (size_t)e * D_DIM;
      v2f A0, A1;
      A0[0] = wie[koff];
      A0[1] = wie[koff + 1];
      A1[0] = half ? 0.0f : wie[4];               // K4: W_inv[e][4]
      A1[1] = half ? 0.0f : bii[e];               // K5: b_inv[e] (x 1.0)
      v8f q = {0,0,0,0,0,0,0,0};
      q = __builtin_amdgcn_wmma_f32_16x16x4_f32(false, A0, false, B0, (short)0, q, false, false);
      q = __builtin_amdgcn_wmma_f32_16x16x4_f32(false, A1, false, B1, (short)0, q, false, false);
      // lane owns 8 consecutive e's of its batch-row
      float* rr = rbuf + row * RSTRIDE + c * 16 + half * 8;
      v4f r0v = *(const v4f*)rr;
      v4f r1v = *(const v4f*)(rr + 4);
      r0v[0] -= q[0]; r0v[1] -= q[1]; r0v[2] -= q[2]; r0v[3] -= q[3];
      r1v[0] -= q[4]; r1v[1] -= q[5]; r1v[2] -= q[6]; r1v[3] -= q[7];
      *(v4f*)rr = r0v;
      *(v4f*)(rr + 4) = r1v;
    }

    snap(i);  // rows whose mixture selects stage i: loss uses ||r after stage i||
  }

  // ---------- q = h_in - r_final ----------
  for (int r = 0; r < ROWS_PER_WAVE; ++r) {
    const float* hsrc = h_in + (size_t)(row0 + r) * E_DIM;
    float* qdst = out_q + (size_t)(row0 + r) * E_DIM;
    const float* rsrc = rbuf + r * RSTRIDE;
#pragma unroll
    for (int c = 0; c < 8; ++c) {
      const int e = c * 128 + lane * 4;
      v4f hv = *(const v4f*)(hsrc + e);
      v4f rv = *(const v4f*)(rsrc + e);
      *(v4f*)(qdst + e) = hv - rv;
    }
  }

  // ---------- per-tile loss partial (deterministic, no atomics) ----------
  float contrib = (lane < 16) ? sqrtf(rowssq) : 0.0f;
  contrib += __shfl_xor(contrib, 1, 32);  contrib += __shfl_xor(contrib, 2, 32);
  contrib += __shfl_xor(contrib, 4, 32);  contrib += __shfl_xor(contrib, 8, 32);
  contrib += __shfl_xor(contrib, 16, 32);
  if (lane == 0) ws[tile] = contrib;
}

__global__ void rvq_loss_reduce(const float* __restrict__ ws, float* __restrict__ out_loss) {
  __shared__ float red[256];
  float a = 0.0f;
  for (int j = threadIdx.x; j < N_TILES; j += 256) a += ws[j];  // fixed order
  red[threadIdx.x] = a;
  __syncthreads();
  for (int s = 128; s > 0; s >>= 1) {
    if ((int)threadIdx.x < s) red[threadIdx.x] += red[threadIdx.x + s];
    __syncthreads();
  }
  if (threadIdx.x == 0) out_loss[0] = red[0] * (1.0f / (float)N_ROWS);
}

extern "C" void kernel_launch(void* const* d_in, const int* in_sizes, int n_in,
                              void* d_out, int out_size, void* d_ws, size_t ws_size,
                              hipStream_t stream) {
  const float* h_in  = (const float*)d_in[0];
  const float* W_prj = (const float*)d_in[1];
  const float* b_prj = (const float*)d_in[2];
  const float* W_inv = (const float*)d_in[3];
  const float* b_inv = (const float*)d_in[4];
  const float* temp  = (const float*)d_in[5];
  const float* u     = (const float*)d_in[6];
  const float* p     = (const float*)d_in[7];

  float* out      = (float*)d_out;
  float* out_q    = out;
  float* out_code = out + Q_ELEMS;
  float* out_loss = out + Q_ELEMS + CODE_ELEMS;
  float* ws       = (float*)d_ws;

  (void)hipFuncSetAttribute((const void*)rvq_fused,
                            hipFuncAttributeMaxDynamicSharedMemorySize, LDS_BYTES);

  rvq_fused<<<N_TILES / WAVES_PER_WG, WAVES_PER_WG * 32, LDS_BYTES, stream>>>(
      h_in, W_prj, b_prj, W_inv, b_inv, temp, u, p, out_q, out_code, ws);
  rvq_loss_reduce<<<1, 256, 0, stream>>>(ws, out_loss);
}